// Glance_6957847020181
// MI455X (gfx1250) — compile-verified
//
#include <hip/hip_runtime.h>
#include <hip/hip_bf16.h>
#include <math.h>

#define ALPHA_C 0.7f
#define SIGMA_C 0.1f
#define MIN_MINING_STEP 50
#define TMAX 2048
#define BLOCK 256
#define CHUNK 32

typedef __attribute__((ext_vector_type(2))) float v2f;
typedef __attribute__((ext_vector_type(8))) float v8f;
typedef __attribute__((ext_vector_type(4))) unsigned int u32x4;
typedef __attribute__((ext_vector_type(8))) int i32x8;
typedef __attribute__((ext_vector_type(4))) int i32x4;

// LDS layout (51200 bytes total):
//  [0,      8192)  s_lds    : float[TMAX]  mean-sigmoid probabilities
//  [8192,  16384)  temp     : float[TMAX]  splat output
//  [16384, 24576)  anc_pos  : float[TMAX]  (aliases TDM staging rows)
//  [24576, 32768)  anc_A1   : float[TMAX]  (aliases TDM staging rows)
//  [32768, 40960)  anc_A2   : float[TMAX]
//  [40960, 43008)  afl      : u8[TMAX]     anchor flags
//  [43008, 45056)  gff      : u8[TMAX]     forward-grown flags
//  [45056, 47104)  gbf      : u8[TMAX]     backward-grown flags
//  [47104, 49152)  red      : float[512]   reductions
//  [49152, 51200)  chunked-scan state (fwd = idx 0..63, bwd = idx 64..127)

__global__ __launch_bounds__(BLOCK)
void glance_row_kernel(const float* __restrict__ scores,
                       const int* __restrict__ plabel,
                       const int* __restrict__ seqlen,
                       const int* __restrict__ step_p,
                       float* __restrict__ partial,
                       int C, int T) {
  __shared__ __align__(16) unsigned char smem[51200];
  float* s_lds   = (float*)(smem);
  float* temp    = (float*)(smem + 8192);
  float* anc_pos = (float*)(smem + 16384);
  float* anc_A1  = (float*)(smem + 24576);
  float* anc_A2  = (float*)(smem + 32768);
  unsigned char* afl = smem + 40960;
  unsigned char* gff = smem + 43008;
  unsigned char* gbf = smem + 45056;
  float* red = (float*)(smem + 47104);
  float* staging = (float*)(smem + 16384);  // C*T raw scores (C=2 -> 16KB)
  // chunked-scan state: 128 slots (0..63 fwd, 64..127 bwd)
  float* cmin2   = (float*)(smem + 49152);
  float* anchA2  = (float*)(smem + 49664);
  float* inanch2 = (float*)(smem + 50176);
  unsigned char* hasA2  = smem + 50688;
  unsigned char* okA2   = smem + 50816;
  unsigned char* inok2  = smem + 50944;
  unsigned char* inhas2 = smem + 51072;
  __shared__ int cnt_sh;
  __shared__ float stat_sh[2];

  const int b   = blockIdx.x;
  const int tid = threadIdx.x;
  const int step = step_p[0];
  const int vl   = seqlen[b];
  const float* srow = scores + (size_t)b * C * T;
  const int nch = T / CHUNK;   // <= 64

  // ---- Stage raw score rows into LDS (TDM DMA on CDNA5, else vector loads) ----
#if __has_builtin(__builtin_amdgcn_tensor_load_to_lds) && __has_builtin(__builtin_amdgcn_s_wait_tensorcnt)
  if (tid < 32) {
    unsigned lds_off = (unsigned)(uintptr_t)(void*)staging;  // low 32 bits = LDS byte addr
    unsigned long long ga = (unsigned long long)(uintptr_t)(const void*)srow;
    u32x4 g0;
    g0[0] = 1u;                                        // count=1, user descriptor
    g0[1] = lds_off;                                   // lds_addr
    g0[2] = (unsigned)(ga & 0xFFFFFFFFu);              // global_addr lo
    g0[3] = (unsigned)((ga >> 32) & 0x01FFFFFFu) | 0x80000000u;  // addr hi | type=2
    i32x8 g1;
    g1[0] = (int)(2u << 16);                           // data_size = 4 bytes
    g1[1] = (int)((unsigned)T << 16);                  // tensor_dim0[15:0] @ bits 63:48
    g1[2] = (int)((((unsigned)T >> 16) & 0xFFFFu) | ((unsigned)C << 16)); // dim0 hi | tensor_dim1 lo
    g1[3] = (int)((unsigned)T << 16);                  // tensor_dim1 hi=0 | tile_dim0 = T
    g1[4] = (int)((unsigned)C & 0xFFFFu);              // tile_dim1 = C, tile_dim2 = 0
    g1[5] = (int)T;                                    // tensor_dim0_stride lo32
    g1[6] = 0;                                         // stride hi | dim1_stride lo
    g1[7] = 0;
    i32x4 z4 = {};
    i32x8 z8 = {};
    __builtin_amdgcn_tensor_load_to_lds(g0, g1, z4, z4, z8, 0);
    __builtin_amdgcn_s_wait_tensorcnt(0);
  }
  __syncthreads();
#else
  for (int i = tid; i < C * T; i += BLOCK) staging[i] = srow[i];
  __syncthreads();
#endif

  // ---- probs = mean_c sigmoid(score), anchor flags ----
  for (int t = tid; t < T; t += BLOCK) {
    float acc = 0.0f;
    for (int c = 0; c < C; ++c) {
      float x = staging[c * T + t];
      acc += 1.0f / (1.0f + expf(-x));
    }
    s_lds[t] = acc / (float)C;
    afl[t] = (plabel[(size_t)b * T + t] > 0) ? 1 : 0;
  }
  if (tid == 0) cnt_sh = 0;
  __syncthreads();

  // =====================================================================
  // Mining scan, 3-pass chunked (removes the 2048-step serial LDS chain).
  // Recurrence per element:  grown = ok && (s >= a*anch) && has;
  //                          ok' = anchor ? true : grown; anch' = anchor ? s : anch;
  // Across an anchor-free chunk: ok_out = ok_in && has_in && (min(s) >= a*anch_in).
  // After the first anchor in a chunk the state is incoming-independent.
  // =====================================================================

  // ---- Pass 1: per-chunk summaries (fwd: tid 0..nch-1, bwd: tid 64..64+nch-1) ----
  if (tid < nch) {
    const int base = tid * CHUNK;
    bool seen = false, ok = true;
    float anch = 0.0f, cmin = INFINITY;
#pragma unroll 4
    for (int i = 0; i < CHUNK; ++i) {
      float sj = s_lds[base + i];
      bool aj = afl[base + i] != 0;
      cmin = fminf(cmin, sj);
      if (aj) { ok = true; anch = sj; seen = true; }
      else if (seen) { ok = ok && (sj >= ALPHA_C * anch); }
    }
    cmin2[tid] = cmin; hasA2[tid] = seen ? 1 : 0;
    okA2[tid] = ok ? 1 : 0; anchA2[tid] = anch;
  } else if (tid >= 64 && tid < 64 + nch) {
    const int c = tid - 64;
    bool seen = false, ok = true;
    float anch = 0.0f, cmin = INFINITY;
#pragma unroll 4
    for (int i = 0; i < CHUNK; ++i) {
      int t = T - 1 - (c * CHUNK + i);
      float sj = s_lds[t];
      bool aj = afl[t] != 0;
      cmin = fminf(cmin, sj);
      if (aj) { ok = true; anch = sj; seen = true; }
      else if (seen) { ok = ok && (sj >= ALPHA_C * anch); }
    }
    cmin2[64 + c] = cmin; hasA2[64 + c] = seen ? 1 : 0;
    okA2[64 + c] = ok ? 1 : 0; anchA2[64 + c] = anch;
  }
  __syncthreads();

  // ---- Pass 2: carry propagation (fwd on wave0 lane0, bwd on wave1 lane0) ----
  if (tid == 0) {
    bool ok = true, has = false; float anch = 0.0f;
    for (int c = 0; c < nch; ++c) {
      inok2[c] = ok ? 1 : 0; inanch2[c] = anch; inhas2[c] = has ? 1 : 0;
      if (hasA2[c]) { ok = okA2[c] != 0; anch = anchA2[c]; has = true; }
      else          { ok = ok && has && (cmin2[c] >= ALPHA_C * anch); }
    }
  } else if (tid == 32) {
    bool ok = true, has = false; float anch = 0.0f;
    for (int c = 0; c < nch; ++c) {
      inok2[64 + c] = ok ? 1 : 0; inanch2[64 + c] = anch; inhas2[64 + c] = has ? 1 : 0;
      if (hasA2[64 + c]) { ok = okA2[64 + c] != 0; anch = anchA2[64 + c]; has = true; }
      else               { ok = ok && has && (cmin2[64 + c] >= ALPHA_C * anch); }
    }
  }
  __syncthreads();

  // ---- Pass 3: exact replay per chunk with known incoming state ----
  if (tid < nch) {
    const int base = tid * CHUNK;
    bool ok = inok2[tid] != 0, has = inhas2[tid] != 0;
    float anch = inanch2[tid];
#pragma unroll 4
    for (int i = 0; i < CHUNK; ++i) {
      int t = base + i;
      float sj = s_lds[t];
      bool aj = afl[t] != 0;
      bool grown = ok && (sj >= ALPHA_C * anch) && has;
      gff[t] = (grown && (t < vl)) ? 1 : 0;
      ok = aj ? true : grown;
      anch = aj ? sj : anch;
      has = has || aj;
    }
  } else if (tid >= 64 && tid < 64 + nch) {
    const int c = tid - 64;
    bool ok = inok2[64 + c] != 0, has = inhas2[64 + c] != 0;
    float anch = inanch2[64 + c];
#pragma unroll 4
    for (int i = 0; i < CHUNK; ++i) {
      int t = T - 1 - (c * CHUNK + i);
      float sj = s_lds[t];
      bool aj = afl[t] != 0;
      bool grown = ok && (sj >= ALPHA_C * anch) && has;
      gbf[t] = grown ? 1 : 0;
      ok = aj ? true : grown;
      anch = aj ? sj : anch;
      has = has || aj;
    }
  }
  __syncthreads();

  // ---- build anchor list with per-anchor analytic row normalization ----
  // W[p,i] = K*exp(-cc*(i-p)^2); row max = K (i=p); row min at dmax=max(p,T-1-p)
  const float Kc = 1.0f / (SIGMA_C * sqrtf(2.0f * (float)M_PI));
  const float cc = (4.0f / ((float)(T - 1) * (float)(T - 1))) / (2.0f * SIGMA_C * SIGMA_C);
  for (int t = tid; t < T; t += BLOCK) {
    int m = (step < MIN_MINING_STEP) ? (int)afl[t] : (int)(afl[t] | gff[t] | gbf[t]);
    if (m) {
      int idx = atomicAdd(&cnt_sh, 1);
      float dmax = (float)((t > T - 1 - t) ? t : (T - 1 - t));
      float wmin = Kc * expf(-cc * dmax * dmax);
      float inv  = 1.0f / (Kc - wmin);
      anc_pos[idx] = (float)t;
      anc_A1[idx]  = Kc * inv;     // Wn = w*A1 - A2, w = exp(-cc*d^2)
      anc_A2[idx]  = wmin * inv;
    }
  }
  __syncthreads();

  // ---- splat: temp[i] = max over anchors (order-independent) ----
  const int cnt = cnt_sh;
  float lmin = INFINITY, lmax = -INFINITY;
  for (int t = tid; t < T; t += BLOCK) {
    float tv = -INFINITY;
    float fi = (float)t;
    for (int j = 0; j < cnt; ++j) {
      float d = fi - anc_pos[j];
      float w = expf(-cc * d * d);
      float v = fmaf(w, anc_A1[j], -anc_A2[j]);
      tv = fmaxf(tv, v);
    }
    temp[t] = tv;
    lmin = fminf(lmin, tv);
    lmax = fmaxf(lmax, tv);
  }
  red[tid] = lmin;
  red[tid + BLOCK] = lmax;
  __syncthreads();
  if (tid == 0) {
    float mn = INFINITY, mx = -INFINITY;
    for (int i = 0; i < BLOCK; ++i) {
      mn = fminf(mn, red[i]);
      mx = fmaxf(mx, red[i + BLOCK]);
    }
    stat_sh[0] = mn; stat_sh[1] = mx;
  }
  __syncthreads();
  const float tmn = stat_sh[0], tmx = stat_sh[1];
  const float inv2 = (tmx > tmn) ? 1.0f / (tmx - tmn) : 0.0f;

  // ---- cross entropy partial sum ----
  float lsum = 0.0f;
  for (int t = tid; t < T; t += BLOCK) {
    float r;
    if (cnt == 0) {
      r = 0.0f;
    } else {
      float tv = temp[t];
      r = (tmx > tmn) ? (tv - tmn) * inv2 : tv;
    }
    float sc = s_lds[t];
    lsum += -(r * logf(sc) + (1.0f - r) * log1pf(-sc));
  }
  __syncthreads();

  // ---- block reduction: route lane partials through the matrix pipe ----
#if __has_builtin(__builtin_amdgcn_wmma_f32_16x16x4_f32)
  {
    // A(16x4) holds the 32 lane partials (2 slots/lane, second = 0);
    // B(4x16) = ones => every D[m][n] = rowsum_m; sum(D) = 16 * sum(A).
    v2f a; a[0] = lsum; a[1] = 0.0f;
    v2f ones; ones[0] = 1.0f; ones[1] = 1.0f;
    v8f cacc = {};
    cacc = __builtin_amdgcn_wmma_f32_16x16x4_f32(
        false, a, false, ones, (short)0, cacc, false, false);
    float cs = cacc[0] + cacc[1] + cacc[2] + cacc[3] +
               cacc[4] + cacc[5] + cacc[6] + cacc[7];
    red[tid] = cs;
    __syncthreads();
    if (tid == 0) {
      float tot = 0.0f;
      for (int i = 0; i < BLOCK; ++i) tot += red[i];
      partial[b] = tot * (1.0f / 16.0f);
    }
  }
#else
  red[tid] = lsum;
  __syncthreads();
  if (tid == 0) {
    float tot = 0.0f;
    for (int i = 0; i < BLOCK; ++i) tot += red[i];
    partial[b] = tot;
  }
#endif
}

__global__ void glance_reduce_kernel(const float* __restrict__ partial,
                                     float* __restrict__ out, int nb, int T) {
  if (threadIdx.x == 0) {
    float tot = 0.0f;
    for (int i = 0; i < nb; ++i) tot += partial[i];
    out[0] = tot / (float)(nb * T);
  }
}

extern "C" void kernel_launch(void* const* d_in, const int* in_sizes, int n_in,
                              void* d_out, int out_size, void* d_ws, size_t ws_size,
                              hipStream_t stream) {
  const float* scores = (const float*)d_in[0];
  // d_in[1] = labels (unused by the reference computation)
  const int* plabel = (const int*)d_in[2];
  const int* seqlen = (const int*)d_in[3];
  const int* step_p = (const int*)d_in[4];

  const int B  = in_sizes[3];               // seqlen is (B,)
  const int T  = in_sizes[2] / B;           // point_label is (B, T)
  const int C  = in_sizes[0] / (B * T);     // scores is (B*C, T)
  const int nb = B / 2;

  float* partial = (float*)d_ws;            // nb floats of scratch

  glance_row_kernel<<<nb, BLOCK, 0, stream>>>(scores, plabel, seqlen, step_p,
                                              partial, C, T);
  glance_reduce_kernel<<<1, 32, 0, stream>>>(partial, (float*)d_out, nb, T);
}